// CouchesintermediairesGNN_12043088298533
// MI455X (gfx1250) — compile-verified
//
#include <hip/hip_runtime.h>

#define THRESH 10.0f

typedef __attribute__((ext_vector_type(2))) float v2f;
typedef __attribute__((ext_vector_type(8))) float v8f;

// ---------------------------------------------------------------------------
// Per-edge MLP: acc[10] = relu(d*W1 + b1) @ W2 + b2  (weights in LDS,
// uniform broadcast reads). Bucket index computed separately.
// ---------------------------------------------------------------------------
__device__ __forceinline__ int compute_bucket(float d)
{
    int bi = (int)(d * (10.0f / THRESH));
    return bi < 0 ? 0 : (bi > 9 ? 9 : bi);
}

__device__ __forceinline__ void compute_mlp(float d,
    const float* __restrict__ sW1, const float* __restrict__ sb1,
    const float* __restrict__ sW2, const float* __restrict__ sb2,
    float* acc)
{
#pragma unroll
    for (int k = 0; k < 10; ++k) acc[k] = sb2[k];
#pragma unroll 4
    for (int j = 0; j < 64; ++j) {
        float hj = fmaxf(fmaf(d, sW1[j], sb1[j]), 0.0f);   // relu hidden
#pragma unroll
        for (int k = 0; k < 10; ++k)
            acc[k] = fmaf(hj, sW2[j * 10 + k], acc[k]);
    }
}

// ---------------------------------------------------------------------------
// Zero-init scratch (sum_w) and output (nbf accumulator region).
// ---------------------------------------------------------------------------
__global__ void zero_kernel(float* __restrict__ ws, float* __restrict__ out,
                            int nws, int nout)
{
    int i = blockIdx.x * blockDim.x + threadIdx.x;
    int stride = gridDim.x * blockDim.x;
    for (int k = i; k < nws; k += stride) ws[k] = 0.0f;
    for (int k = i; k < nout; k += stride) out[k] = 0.0f;
}

// ---------------------------------------------------------------------------
// Edge pass 1: sum_w[src] += comb. One-hot half contributes exactly one
// nonzero -> 11 atomics per edge instead of 20 (L2-resident 8MB table).
// ---------------------------------------------------------------------------
__global__ void edge_pass1(const int* __restrict__ src,
                           const float* __restrict__ ea,
                           const float* __restrict__ W1, const float* __restrict__ b1,
                           const float* __restrict__ W2, const float* __restrict__ b2,
                           float* __restrict__ sum_w, int nE)
{
    __shared__ float sW1[64], sb1[64], sW2[640], sb2[16];
    for (int i = threadIdx.x; i < 64; i += blockDim.x) { sW1[i] = W1[i]; sb1[i] = b1[i]; }
    for (int i = threadIdx.x; i < 640; i += blockDim.x) sW2[i] = W2[i];
    if (threadIdx.x < 10) sb2[threadIdx.x] = b2[threadIdx.x];
    __syncthreads();

    int e = blockIdx.x * blockDim.x + threadIdx.x;
    if (e >= nE) return;

    int   s = src[e];
    float d = ea[e];
    int   bi = compute_bucket(d);
    float mlp[10];
    compute_mlp(d, sW1, sb1, sW2, sb2, mlp);

    float* p = sum_w + (size_t)s * 20;
    atomicAdd(p + bi, 1.0f);                 // the whole one-hot contribution
#pragma unroll
    for (int j = 0; j < 10; ++j) atomicAdd(p + 10 + j, mlp[j]);
}

// ---------------------------------------------------------------------------
// Edge pass 2: nbf[src] += |a*h[src] - (1-a)*h[dst]|^b * w_tilde
// nbf accumulates straight into d_out[:,1,:] (node*40 + 20 .. 39).
// comb is recomputed (VALU is free; avoids a 256MB comb spill to HBM).
// Zero-valued contributions skip the atomic (numerically identical).
// ---------------------------------------------------------------------------
__global__ void edge_pass2(const int* __restrict__ src,
                           const int* __restrict__ dst,
                           const float* __restrict__ ea,
                           const float* __restrict__ x,     // (N,2,20): h = x[n*40 + 0..19]
                           const float* __restrict__ W1, const float* __restrict__ b1,
                           const float* __restrict__ W2, const float* __restrict__ b2,
                           const float* __restrict__ a_ptr,
                           const float* __restrict__ b_ptr,
                           const float* __restrict__ sum_w,
                           float* __restrict__ out, int nE)
{
    __shared__ float sW1[64], sb1[64], sW2[640], sb2[16];
    for (int i = threadIdx.x; i < 64; i += blockDim.x) { sW1[i] = W1[i]; sb1[i] = b1[i]; }
    for (int i = threadIdx.x; i < 640; i += blockDim.x) sW2[i] = W2[i];
    if (threadIdx.x < 10) sb2[threadIdx.x] = b2[threadIdx.x];
    __syncthreads();

    int e = blockIdx.x * blockDim.x + threadIdx.x;
    if (e >= nE) return;

    int   s = src[e];
    int   t = dst[e];
    float d = ea[e];

    int   bi = compute_bucket(d);
    float mlp[10];
    compute_mlp(d, sW1, sb1, sW2, sb2, mlp);

    // 16B-aligned b128 gathers of the 20-float rows (tables are L2 resident).
    float sw[20], hs[20], hd[20];
    {
        const float4* swp = (const float4*)(sum_w + (size_t)s * 20);
        const float4* hsp = (const float4*)(x + (size_t)s * 40);
        const float4* hdp = (const float4*)(x + (size_t)t * 40);
#pragma unroll
        for (int q = 0; q < 5; ++q) {
            float4 v0 = swp[q]; float4 v1 = hsp[q]; float4 v2 = hdp[q];
            sw[q*4+0] = v0.x; sw[q*4+1] = v0.y; sw[q*4+2] = v0.z; sw[q*4+3] = v0.w;
            hs[q*4+0] = v1.x; hs[q*4+1] = v1.y; hs[q*4+2] = v1.z; hs[q*4+3] = v1.w;
            hd[q*4+0] = v2.x; hd[q*4+1] = v2.y; hd[q*4+2] = v2.z; hd[q*4+3] = v2.w;
        }
    }

    float av  = a_ptr[0];
    float bv  = b_ptr[0];
    float oma = 1.0f - av;

    float* nbf = out + (size_t)s * 40 + 20;
#pragma unroll
    for (int j = 0; j < 20; ++j) {
        float cj  = (j < 10) ? ((j == bi) ? 1.0f : 0.0f) : mlp[j - 10];
        float wt  = (sw[j] != 0.0f) ? (cj / sw[j]) : 0.01f;
        float mag = fabsf(av * hs[j] - oma * hd[j]);
        float rho = __expf(bv * __logf(mag));   // mag==0 -> rho==0 (b>1)
        float val = rho * wt;
        if (val != 0.0f) atomicAdd(nbf + j, val);
    }
}

// ---------------------------------------------------------------------------
// Node update with V_WMMA_F32_16X16X4_F32:
//   new_h = sigmoid([h | nbf](N x 40) @ [gamma1^T; gamma2^T](40 x 20) + bias)
// B is staged TRANSPOSED in LDS as Bt[col][k] (stride 44 dwords):
//   - the two K-elements of each B operand are contiguous -> one 8B-aligned
//     ds_load_b64 straight into an even VGPR pair (no repack movs),
//   - gcd(44,64)==4 makes the full-wave b64 read touch all 64 banks exactly
//     once (conflict-free).
// A fetches are 8B-aligned global_load_b64 (base pointer selected on the
// compile-time unrolled k0, no divergence). Tile-granular exit keeps EXEC
// all-ones for every WMMA.
// ---------------------------------------------------------------------------
#define BT_STRIDE 44

__global__ void node_update_wmma(const float* __restrict__ x,
                                 const float* __restrict__ g1,
                                 const float* __restrict__ g2,
                                 const float* __restrict__ bias,
                                 float* __restrict__ out, int n)
{
    __shared__ float Bt[32][BT_STRIDE];     // Bt[col][k]; cols >= 20 are zero
    for (int idx = threadIdx.x; idx < 32 * 40; idx += blockDim.x) {
        int c = idx / 40, k = idx % 40;
        float v = 0.0f;
        if (c < 20) v = (k < 20) ? g1[c * 20 + k] : g2[c * 20 + (k - 20)];
        Bt[c][k] = v;
    }
    __syncthreads();

    int wave = blockIdx.x * (blockDim.x >> 5) + (threadIdx.x >> 5);
    int lane = threadIdx.x & 31;
    int row0 = wave * 16;
    if (row0 >= n) return;                  // uniform per wave -> EXEC stays full

    int  m    = lane & 15;
    bool hi   = lane >= 16;
    int  node = row0 + m;
    int  col0 = m;                          // N-tile 0 column
    int  col1 = 16 + m;                     // N-tile 1 column (valid if < 20)

    const float* xrow = x   + (size_t)node * 40;  // k in [0,20): h
    const float* orow = out + (size_t)node * 40;  // k in [20,40): nbf

    v8f c0 = {};
    v8f c1 = {};

#pragma unroll
    for (int k0 = 0; k0 < 40; k0 += 4) {
        // For every lane in this (unrolled) iteration: ka,kb < 20 iff k0 < 20.
        const float* base = (k0 < 20) ? xrow : orow;
        int ka = k0 + (hi ? 2 : 0);         // kb = ka + 1 (adjacent, 8B aligned)
        float2 av2 = *(const float2*)(base + ka);
        float2 bv0 = *(const float2*)(&Bt[col0][ka]);   // one ds_load_b64
        float2 bv1 = *(const float2*)(&Bt[col1][ka]);   // one ds_load_b64
        v2f a;   a.x   = av2.x;  a.y   = av2.y;
        v2f bm0; bm0.x = bv0.x;  bm0.y = bv0.y;
        v2f bm1; bm1.x = bv1.x;  bm1.y = bv1.y;
        c0 = __builtin_amdgcn_wmma_f32_16x16x4_f32(
                 false, a, false, bm0, (short)0, c0, false, false);
        c1 = __builtin_amdgcn_wmma_f32_16x16x4_f32(
                 false, a, false, bm1, (short)0, c1, false, false);
    }

    float bia0 = bias[col0];
    float bia1 = (col1 < 20) ? bias[col1] : 0.0f;

#pragma unroll
    for (int v = 0; v < 8; ++v) {
        int mrow  = v + (hi ? 8 : 0);
        int onode = row0 + mrow;
        float s0 = c0[v] + bia0;
        out[(size_t)onode * 40 + col0] = 1.0f / (1.0f + __expf(-s0));
        if (col1 < 20) {
            float s1 = c1[v] + bia1;
            out[(size_t)onode * 40 + col1] = 1.0f / (1.0f + __expf(-s1));
        }
    }
}

// ---------------------------------------------------------------------------
extern "C" void kernel_launch(void* const* d_in, const int* in_sizes, int n_in,
                              void* d_out, int out_size, void* d_ws, size_t ws_size,
                              hipStream_t stream)
{
    const float* x    = (const float*)d_in[0];   // (N,2,20)
    const int*   ei   = (const int*)  d_in[1];   // (2,E)
    const float* ea   = (const float*)d_in[2];   // (E,1)
    const float* W1   = (const float*)d_in[3];   // (1,64)
    const float* b1   = (const float*)d_in[4];   // (64,)
    const float* W2   = (const float*)d_in[5];   // (64,10)
    const float* b2   = (const float*)d_in[6];   // (10,)
    const float* a    = (const float*)d_in[7];   // (1,)
    const float* b    = (const float*)d_in[8];   // (1,)
    const float* g1   = (const float*)d_in[9];   // (20,20)
    const float* g2   = (const float*)d_in[10];  // (20,20)
    const float* bias = (const float*)d_in[11];  // (20,)

    const int E_ = in_sizes[1] / 2;
    const int N_ = in_sizes[0] / 40;

    const int* src = ei;
    const int* dst = ei + E_;

    float* sum_w = (float*)d_ws;                 // N*20 floats of scratch
    float* out   = (float*)d_out;                // (N,2,20)

    (void)n_in; (void)out_size; (void)ws_size;

    // 1) zero sum_w and the whole output (nbf region must start at 0)
    {
        int nws = N_ * 20, nout = N_ * 40;
        zero_kernel<<<1024, 256, 0, stream>>>(sum_w, out, nws, nout);
    }
    // 2) edge pass 1: build sum_w
    {
        int blocks = (E_ + 255) / 256;
        edge_pass1<<<blocks, 256, 0, stream>>>(src, ea, W1, b1, W2, b2, sum_w, E_);
    }
    // 3) edge pass 2: accumulate nbf into out[:,1,:]
    {
        int blocks = (E_ + 255) / 256;
        edge_pass2<<<blocks, 256, 0, stream>>>(src, dst, ea, x, W1, b1, W2, b2,
                                               a, b, sum_w, out, E_);
    }
    // 4) node update via WMMA: writes new_h into out[:,0,:]
    {
        int waves  = (N_ + 15) / 16;
        int blocks = (waves + 7) / 8;            // 8 waves (256 threads) per block
        node_update_wmma<<<blocks, 256, 0, stream>>>(x, g1, g2, bias, out, N_);
    }
}